// GNN_46033459479308
// MI455X (gfx1250) — compile-verified
//
#include <hip/hip_runtime.h>
#include <hip/hip_bf16.h>

typedef __attribute__((ext_vector_type(2))) float v2f;
typedef __attribute__((ext_vector_type(8))) float v8f;
typedef __attribute__((ext_vector_type(4))) unsigned int v4u;
typedef __attribute__((ext_vector_type(8))) int v8i;
typedef __attribute__((ext_vector_type(4))) int v4i;

#define LN_EPS 1e-5f

#if defined(__has_builtin)
#if __has_builtin(__builtin_amdgcn_tensor_load_to_lds) && __has_builtin(__builtin_amdgcn_s_wait_tensorcnt)
#define USE_TDM 1
#endif
#endif

// ---------------------------------------------------------------------------
// Generic fp32 WMMA GEMM: C[M,Ncols] = A[M,K] @ W[K,Ncols] + bias (+resid)(+relu)
// Block = 4 waves x 16 rows = 64 rows; each wave computes a 16x32 tile (two
// V_WMMA_F32_16X16X4_F32 accumulators sharing one A fragment).
// The W panel (K x 32) is staged into LDS once per block via the Tensor Data
// Mover (tensor_load_to_lds + s_wait_tensorcnt) when available.
// Grid: x = ceil(M/64), y = Ncols/32. Dynamic LDS = K*32*4 bytes.
// ---------------------------------------------------------------------------
extern __shared__ float Wlds[];   // [K][32], row-major

__global__ __launch_bounds__(128) void gemm_f32_wmma(
    const float* __restrict__ A, int lda,
    const float* __restrict__ W, int ldw,
    const float* __restrict__ bias,
    const float* __restrict__ resid, int ldr,
    float* __restrict__ C, int ldc, int coff,
    int M, int K, int doRelu)
{
    const int wave = threadIdx.x >> 5;
    const int lane = threadIdx.x & 31;
    const int n0 = blockIdx.y * 32;

    // ---- stage W[0:K, n0:n0+32] into LDS ----
#ifdef USE_TDM
    if (wave == 0) {
        // Build Tensor DMA descriptor (D#) groups per CDNA5 ISA 8.3-8.6.
        const unsigned long long gaddr =
            (unsigned long long)(uintptr_t)(W + n0);            // tile start
        const unsigned int ldsoff = (unsigned int)(uintptr_t)Wlds; // addr[31:0] = LDS offset

        unsigned long long p[2];
        p[0] = 1ull                                             // count=1 (valid, user mode)
             | ((unsigned long long)ldsoff << 32);              // lds_addr [63:32]
        p[1] = (gaddr & ((1ull << 57) - 1ull))                  // global_addr [120:64]
             | (2ull << 62);                                    // type=2 ("image") [127:126]

        unsigned long long q[4] = {0ull, 0ull, 0ull, 0ull};
        // helper: set bits [pos, pos+width) of the 256-bit group1
        #define SETF(pos, val) do {                                        \
            const int _w = (pos) >> 6, _b = (pos) & 63;                    \
            const unsigned long long _v = (unsigned long long)(val);       \
            q[_w] |= _v << _b;                                             \
            if (_b) q[_w + 1] |= _v >> (64 - _b);                          \
        } while (0)
        SETF(16, 2ull);                         // data_size = 2 -> 4 bytes
        SETF(48, (unsigned long long)(unsigned)ldw);   // tensor_dim0
        SETF(80, (unsigned long long)(unsigned)K);     // tensor_dim1
        SETF(112, 32ull);                       // tile_dim0 = 32 elements
        SETF(128, (unsigned long long)(unsigned)K);    // tile_dim1 = K rows
        SETF(160, (unsigned long long)(unsigned)ldw);  // tensor_dim0_stride
        #undef SETF

        v4u g0; __builtin_memcpy(&g0, p, 16);
        v8i g1; __builtin_memcpy(&g1, q, 32);
        v4i g2 = {0, 0, 0, 0};
        v4i g3 = {0, 0, 0, 0};
#if __clang_major__ >= 23
        v8i g4 = {0, 0, 0, 0, 0, 0, 0, 0};
        __builtin_amdgcn_tensor_load_to_lds(g0, g1, g2, g3, g4, 0);
#else
        __builtin_amdgcn_tensor_load_to_lds(g0, g1, g2, g3, 0);
#endif
        __builtin_amdgcn_s_wait_tensorcnt(0);
    }
#else
    for (int c = threadIdx.x; c < K * 8; c += 128) {   // 8 float4 per K-row
        const int k = c >> 3, qd = c & 7;
        const float4 v = *(const float4*)(W + (size_t)k * ldw + n0 + qd * 4);
        *(float4*)(Wlds + k * 32 + qd * 4) = v;
    }
#endif
    __syncthreads();

    const int m0 = (blockIdx.x * 4 + wave) * 16;
    if (m0 >= M) return;                    // wave-uniform; EXEC all-1 for WMMA

    const int half = lane >> 4;             // 0 or 1
    const int l15  = lane & 15;
    const int rowA = m0 + l15;
    const float* __restrict__ aRow = A + (size_t)rowA * lda;
    __builtin_prefetch(aRow, 0, 0);         // global_prefetch_b8 (A row, L2)

    v8f acc0 = {}, acc1 = {};
    for (int k0 = 0; k0 < K; k0 += 4) {
        const int kk = k0 + (half << 1);    // K-pair handled by this lane half
        const v2f a = *(const v2f*)(aRow + kk);     // b64 load
        v2f b0, b1;
        b0.x = Wlds[kk * 32 + l15];
        b0.y = Wlds[(kk + 1) * 32 + l15];
        b1.x = Wlds[kk * 32 + 16 + l15];
        b1.y = Wlds[(kk + 1) * 32 + 16 + l15];
        acc0 = __builtin_amdgcn_wmma_f32_16x16x4_f32(
            false, a, false, b0, (short)0, acc0, false, false);
        acc1 = __builtin_amdgcn_wmma_f32_16x16x4_f32(
            false, a, false, b1, (short)0, acc1, false, false);
    }

    // C/D layout: col = base + (lane&15); VGPR i -> row m0 + (lane>>4)*8 + i
    const int col0 = n0 + l15;
    const int col1 = col0 + 16;
    const float bv0 = bias ? bias[col0] : 0.0f;
    const float bv1 = bias ? bias[col1] : 0.0f;
    const int rowBase = m0 + (half << 3);
#pragma unroll
    for (int i = 0; i < 8; ++i) {
        const int row = rowBase + i;
        float v0 = acc0[i] + bv0;
        float v1 = acc1[i] + bv1;
        if (resid) {
            v0 += resid[(size_t)row * ldr + col0];
            v1 += resid[(size_t)row * ldr + col1];
        }
        if (doRelu) {
            v0 = v0 > 0.0f ? v0 : 0.0f;
            v1 = v1 > 0.0f ? v1 : 0.0f;
        }
        C[(size_t)row * ldc + col0 + coff] = v0;
        C[(size_t)row * ldc + col1 + coff] = v1;
    }
}

// ---------------------------------------------------------------------------
// Row-wise LayerNorm (+optional ReLU). One wave per row, D in {128, 256}.
// ---------------------------------------------------------------------------
__global__ __launch_bounds__(256) void ln_relu_kernel(
    const float* __restrict__ X, float* __restrict__ Y,
    const float* __restrict__ gamma, const float* __restrict__ beta,
    int D, int M, int doRelu)
{
    const int wave = threadIdx.x >> 5;
    const int lane = threadIdx.x & 31;
    const int row = blockIdx.x * 8 + wave;
    if (row >= M) return;

    const float4* __restrict__ x4 = (const float4*)(X + (size_t)row * D);
    float4* __restrict__ y4 = (float4*)(Y + (size_t)row * D);
    const int nv = D >> 7;                       // float4s per lane: 1 or 2

    float4 vals[2];
    float s = 0.0f, ss = 0.0f;
#pragma unroll
    for (int i = 0; i < 2; ++i) {
        if (i < nv) {
            float4 v = x4[lane + i * 32];
            vals[i] = v;
            s  += v.x + v.y + v.z + v.w;
            ss += v.x * v.x + v.y * v.y + v.z * v.z + v.w * v.w;
        }
    }
#pragma unroll
    for (int off = 16; off > 0; off >>= 1) {     // wave32 reduction
        s  += __shfl_xor(s,  off, 32);
        ss += __shfl_xor(ss, off, 32);
    }
    const float inv = 1.0f / (float)D;
    const float mean = s * inv;
    const float var  = ss * inv - mean * mean;
    const float rstd = rsqrtf(var + LN_EPS);

#pragma unroll
    for (int i = 0; i < 2; ++i) {
        if (i < nv) {
            float4 v = vals[i];
            const float4 g = ((const float4*)gamma)[lane + i * 32];
            const float4 b = ((const float4*)beta)[lane + i * 32];
            v.x = (v.x - mean) * rstd * g.x + b.x;
            v.y = (v.y - mean) * rstd * g.y + b.y;
            v.z = (v.z - mean) * rstd * g.z + b.z;
            v.w = (v.w - mean) * rstd * g.w + b.w;
            if (doRelu) {
                v.x = v.x > 0.f ? v.x : 0.f; v.y = v.y > 0.f ? v.y : 0.f;
                v.z = v.z > 0.f ? v.z : 0.f; v.w = v.w > 0.f ? v.w : 0.f;
            }
            y4[lane + i * 32] = v;
        }
    }
}

// ---------------------------------------------------------------------------
// Edge scatter: aggr[dst] += ew * h[src]. One wave per edge (128 floats).
// h (25.6 MB) fits in the 192 MB L2, so gathers are L2-served.
// ---------------------------------------------------------------------------
__global__ __launch_bounds__(256) void edge_scatter(
    const long long* __restrict__ ei, const float* __restrict__ ew,
    const float* __restrict__ h, float* __restrict__ aggr, int E)
{
    const int wave = threadIdx.x >> 5;
    const int lane = threadIdx.x & 31;
    const long long e = (long long)blockIdx.x * 8 + wave;
    if (e >= E) return;
    const int s = (int)ei[e];        // edge_index[0] = src
    const int d = (int)ei[E + e];    // edge_index[1] = dst
    const float w = ew[e];
    const float4 v = ((const float4*)(h + (size_t)s * 128))[lane];
    float* out = aggr + (size_t)d * 128 + lane * 4;
    atomicAdd(out + 0, w * v.x);
    atomicAdd(out + 1, w * v.y);
    atomicAdd(out + 2, w * v.z);
    atomicAdd(out + 3, w * v.w);
}

// ---------------------------------------------------------------------------
// Masked global-add pool: gs[batch[n]] += hs[n] if mask[n]. One wave per node.
// ---------------------------------------------------------------------------
__global__ __launch_bounds__(256) void mask_pool(
    const float* __restrict__ hs, const unsigned char* __restrict__ mask,
    const long long* __restrict__ batch, float* __restrict__ gs, int M)
{
    const int wave = threadIdx.x >> 5;
    const int lane = threadIdx.x & 31;
    const int n = blockIdx.x * 8 + wave;
    if (n >= M) return;
    if (!mask[n]) return;
    const int g = (int)batch[n];
    const float4* __restrict__ src = (const float4*)(hs + (size_t)n * 384);
    float* __restrict__ out = gs + (size_t)g * 384;
#pragma unroll
    for (int i = 0; i < 3; ++i) {
        const int idx = lane + i * 32;
        const float4 v = src[idx];
        atomicAdd(out + idx * 4 + 0, v.x);
        atomicAdd(out + idx * 4 + 1, v.y);
        atomicAdd(out + idx * 4 + 2, v.z);
        atomicAdd(out + idx * 4 + 3, v.w);
    }
}

__global__ __launch_bounds__(256) void zero_f32(float* __restrict__ p, long long n4)
{
    const long long i = (long long)blockIdx.x * 256 + threadIdx.x;
    if (i < n4) ((float4*)p)[i] = make_float4(0.f, 0.f, 0.f, 0.f);
}

// ---------------------------------------------------------------------------
// Orchestration
// ---------------------------------------------------------------------------
static inline void run_gemm(hipStream_t stream,
                            const float* A, int lda, const float* W, int Ncols,
                            const float* bias, const float* resid, int ldr,
                            float* C, int ldc, int coff, int M, int K, int relu)
{
    dim3 grid((M + 63) / 64, Ncols / 32);
    const size_t shmem = (size_t)K * 32 * sizeof(float);
    gemm_f32_wmma<<<grid, 128, shmem, stream>>>(A, lda, W, Ncols, bias, resid, ldr,
                                                C, ldc, coff, M, K, relu);
}

static inline void run_zero(hipStream_t stream, float* p, long long nElems)
{
    long long n4 = nElems / 4;
    zero_f32<<<(unsigned)((n4 + 255) / 256), 256, 0, stream>>>(p, n4);
}

extern "C" void kernel_launch(void* const* d_in, const int* in_sizes, int n_in,
                              void* d_out, int out_size, void* d_ws, size_t ws_size,
                              hipStream_t stream)
{
    const int HID = 128, NL = 3, G = 256;
    const int N = in_sizes[0] / HID;      // 50000
    const int E = in_sizes[2];            // 800000

    // ---- inputs (setup_inputs dict insertion order, params flattened) ----
    const float*      x     = (const float*)d_in[0];
    const long long*  ei    = (const long long*)d_in[1];
    const float*      ew    = (const float*)d_in[2];
    const long long*  batch = (const long long*)d_in[3];
    const unsigned char* mask = (const unsigned char*)d_in[4];
    const float* lin_w = (const float*)d_in[5];
    const float* lin_b = (const float*)d_in[6];
    const int LBASE = 7, NHB = 31, GHB = 39;

    // ---- workspace layout (floats) ----
    float* ws = (float*)d_ws;
    size_t off = 0;
    float* h_in  = ws + off; off += (size_t)N * HID;
    float* hbuf0 = ws + off; off += (size_t)N * HID;
    float* hbuf1 = ws + off; off += (size_t)N * HID;
    float* hbuf2 = ws + off; off += (size_t)N * HID;
    float* aggr  = ws + off; off += (size_t)N * HID;
    float* t256a = ws + off; off += (size_t)N * 2 * HID;
    float* t256b = ws + off; off += (size_t)N * 2 * HID;
    float* t128  = ws + off; off += (size_t)N * HID;
    float* gs    = ws + off; off += (size_t)G * (NL * HID);
    float* gt1   = ws + off; off += (size_t)G * 768;
    float* gt2   = ws + off; off += (size_t)G * 768;
    float* gt3   = ws + off; off += (size_t)G * 384;
    (void)ws_size; (void)n_in; (void)out_size;

    float* hbufs[3] = {hbuf0, hbuf1, hbuf2};
    float* out_hs = (float*)d_out;                        // [N, 384]
    float* out_g  = (float*)d_out + (size_t)N * NL * HID; // [256, 384]

    // ---- h = x @ lin_w + lin_b ----
    run_gemm(stream, x, 128, lin_w, HID, lin_b, nullptr, 0, h_in, HID, 0, N, 128, 0);

    // ---- GIN layers ----
    const float* h_prev = h_in;
    for (int l = 0; l < NL; ++l) {
        const float* w1   = (const float*)d_in[LBASE + l * 8 + 0];
        const float* b1   = (const float*)d_in[LBASE + l * 8 + 1];
        const float* ln_g = (const float*)d_in[LBASE + l * 8 + 2];
        const float* ln_b = (const float*)d_in[LBASE + l * 8 + 3];
        const float* w2   = (const float*)d_in[LBASE + l * 8 + 4];
        const float* b2   = (const float*)d_in[LBASE + l * 8 + 5];
        const float* ng   = (const float*)d_in[LBASE + l * 8 + 6];
        const float* nb   = (const float*)d_in[LBASE + l * 8 + 7];

        run_zero(stream, aggr, (long long)N * HID);
        edge_scatter<<<(E + 7) / 8, 256, 0, stream>>>(ei, ew, h_prev, aggr, E);

        run_gemm(stream, aggr, HID, w1, 2 * HID, b1, nullptr, 0, t256a, 2 * HID, 0, N, HID, 0);
        ln_relu_kernel<<<(N + 7) / 8, 256, 0, stream>>>(t256a, t256a, ln_g, ln_b, 2 * HID, N, 1);

        run_gemm(stream, t256a, 2 * HID, w2, HID, b2, aggr, HID, t128, HID, 0, N, 2 * HID, 0);
        ln_relu_kernel<<<(N + 7) / 8, 256, 0, stream>>>(t128, hbufs[l], ng, nb, HID, N, 1);
        h_prev = hbufs[l];
    }

    // ---- node head FFN per layer, output into d_out[:, l*128:(l+1)*128] ----
    {
        const float* w1 = (const float*)d_in[NHB + 0]; const float* b1 = (const float*)d_in[NHB + 1];
        const float* w2 = (const float*)d_in[NHB + 2]; const float* b2 = (const float*)d_in[NHB + 3];
        const float* w3 = (const float*)d_in[NHB + 4]; const float* b3 = (const float*)d_in[NHB + 5];
        const float* wsk = (const float*)d_in[NHB + 6]; const float* bs = (const float*)d_in[NHB + 7];
        for (int l = 0; l < NL; ++l) {
            run_gemm(stream, hbufs[l], HID, w1, 2 * HID, b1, nullptr, 0, t256a, 2 * HID, 0, N, HID, 1);
            run_gemm(stream, t256a, 2 * HID, w2, 2 * HID, b2, nullptr, 0, t256b, 2 * HID, 0, N, 2 * HID, 1);
            run_gemm(stream, t256b, 2 * HID, w3, HID, b3, nullptr, 0, t128, HID, 0, N, 2 * HID, 1);
            run_gemm(stream, hbufs[l], HID, wsk, HID, bs, t128, HID,
                     out_hs, NL * HID, l * HID, N, HID, 0);
        }
    }

    // ---- masked pooling into gs [256, 384] ----
    run_zero(stream, gs, (long long)G * NL * HID);
    mask_pool<<<(N + 7) / 8, 256, 0, stream>>>(out_hs, mask, batch, gs, N);

    // ---- graph head FFN on [256, 384] ----
    {
        const float* w1 = (const float*)d_in[GHB + 0]; const float* b1 = (const float*)d_in[GHB + 1];
        const float* w2 = (const float*)d_in[GHB + 2]; const float* b2 = (const float*)d_in[GHB + 3];
        const float* w3 = (const float*)d_in[GHB + 4]; const float* b3 = (const float*)d_in[GHB + 5];
        const float* wsk = (const float*)d_in[GHB + 6]; const float* bs = (const float*)d_in[GHB + 7];
        run_gemm(stream, gs, 384, w1, 768, b1, nullptr, 0, gt1, 768, 0, G, 384, 1);
        run_gemm(stream, gt1, 768, w2, 768, b2, nullptr, 0, gt2, 768, 0, G, 768, 1);
        run_gemm(stream, gt2, 768, w3, 384, b3, nullptr, 0, gt3, 384, 0, G, 768, 1);
        run_gemm(stream, gs, 384, wsk, 384, bs, gt3, 384, out_g, 384, 0, G, 384, 0);
    }
}